// HierarchicalPlayerEncoder_21509196218764
// MI455X (gfx1250) — compile-verified
//
#include <hip/hip_runtime.h>
#include <hip/hip_bf16.h>

typedef __attribute__((ext_vector_type(2))) float v2f;
typedef __attribute__((ext_vector_type(8))) float v8f;

#define N_ROWS   131072
#define HID      1024
#define MT       48      // rows per workgroup = 3 WMMA m-tiles
#define HSTRIDE  1028    // f32; 16B row pad -> 4-bank lane stride, conflict-free b64 A-frag loads
#define XSTRIDE  260     // f32; same padding property (1040B % 256 == 16)
#define NWAVE    8
#define BLOCK    256

__device__ __forceinline__ float gelu_exact(float x) {
    // jax.nn.gelu(approximate=False): x * 0.5 * (1 + erf(x / sqrt(2)))
    return 0.5f * x * (1.0f + erff(x * 0.70710678118654752f));
}

// One MLP path for a 48-row tile: h = gelu(LN(x @ W1 + b1)); out_rows(sel) = h @ W2 + b2
template <int K1, bool KNOWN>
__device__ void run_path(float* __restrict__ lds_h, const float* __restrict__ lds_x,
                         float* __restrict__ lds_stats, const int* __restrict__ lds_pid,
                         const float* __restrict__ W1, const float* __restrict__ b1,
                         const float* __restrict__ g,  const float* __restrict__ beta,
                         const float* __restrict__ W2, const float* __restrict__ b2,
                         float* __restrict__ out, long wgBase)
{
    const int tid    = threadIdx.x;
    const int wave   = tid >> 5;
    const int lane   = tid & 31;
    const int laneLo = lane & 15;
    const int kHi    = (lane >> 4) << 1;  // 0 (lanes 0-15) or 2 (lanes 16-31)
    const int hi8    = (lane >> 4) << 3;  // 0 or 8 (C/D high-half row offset)

    // ---------------- GEMM1: lds_h = lds_x @ W1 + b1 (pre-LN) ----------------
    for (int nt = 0; nt < 8; ++nt) {
        const int nb   = wave * 128 + nt * 16;
        const int colB = nb + laneLo;
        const float bias = b1[colB];
        v8f acc[3];
        #pragma unroll
        for (int mt = 0; mt < 3; ++mt)
            #pragma unroll
            for (int r = 0; r < 8; ++r) acc[mt][r] = bias;

        #pragma unroll 4
        for (int k = 0; k < K1; k += 4) {
            const int kk = k + kHi;
            v2f bf;
            bf.x = W1[(size_t)kk * HID + colB];
            bf.y = W1[(size_t)(kk + 1) * HID + colB];
            #pragma unroll
            for (int mt = 0; mt < 3; ++mt) {
                const float* xp = lds_x + (mt * 16 + laneLo) * XSTRIDE + kk;
                v2f af; af.x = xp[0]; af.y = xp[1];
                acc[mt] = __builtin_amdgcn_wmma_f32_16x16x4_f32(
                    false, af, false, bf, (short)0, acc[mt], false, false);
            }
        }
        #pragma unroll
        for (int mt = 0; mt < 3; ++mt)
            #pragma unroll
            for (int r = 0; r < 8; ++r)
                lds_h[(mt * 16 + r + hi8) * HSTRIDE + nb + laneLo] = acc[mt][r];
    }
    __syncthreads();

    // ---------------- per-row mean / rstd (wave butterfly) ----------------
    for (int i = 0; i < MT / NWAVE; ++i) {
        const int row = wave * (MT / NWAVE) + i;
        const float* hp = lds_h + row * HSTRIDE;
        float s = 0.0f, ss = 0.0f;
        for (int c = lane; c < HID; c += 32) { float v = hp[c]; s += v; ss += v * v; }
        #pragma unroll
        for (int off = 16; off > 0; off >>= 1) {
            s  += __shfl_xor(s,  off, 32);
            ss += __shfl_xor(ss, off, 32);
        }
        if (lane == 0) {
            float mu  = s * (1.0f / HID);
            float var = ss * (1.0f / HID) - mu * mu;
            lds_stats[row * 2]     = mu;
            lds_stats[row * 2 + 1] = rsqrtf(var + 1e-5f);
        }
    }
    __syncthreads();

    // ---------------- LayerNorm + exact GELU, in place ----------------
    for (int i = tid; i < MT * HID; i += BLOCK) {
        const int row = i >> 10;
        const int col = i & (HID - 1);
        float v = lds_h[row * HSTRIDE + col];
        v = (v - lds_stats[row * 2]) * lds_stats[row * 2 + 1] * g[col] + beta[col];
        lds_h[row * HSTRIDE + col] = gelu_exact(v);
    }
    __syncthreads();

    // ---------------- GEMM2 + bias + predicated store ----------------
    for (int nt = 0; nt < 8; ++nt) {
        const int nb   = wave * 128 + nt * 16;
        const int colB = nb + laneLo;
        v8f acc[3];
        #pragma unroll
        for (int mt = 0; mt < 3; ++mt)
            #pragma unroll
            for (int r = 0; r < 8; ++r) acc[mt][r] = 0.0f;

        #pragma unroll 4
        for (int k = 0; k < HID; k += 4) {
            const int kk = k + kHi;
            v2f bf;
            bf.x = W2[(size_t)kk * HID + colB];
            bf.y = W2[(size_t)(kk + 1) * HID + colB];
            #pragma unroll
            for (int mt = 0; mt < 3; ++mt) {
                const float* hp = lds_h + (mt * 16 + laneLo) * HSTRIDE + kk;
                v2f af; af.x = hp[0]; af.y = hp[1];
                acc[mt] = __builtin_amdgcn_wmma_f32_16x16x4_f32(
                    false, af, false, bf, (short)0, acc[mt], false, false);
            }
        }
        const float bias = b2[colB];
        #pragma unroll
        for (int mt = 0; mt < 3; ++mt) {
            #pragma unroll
            for (int r = 0; r < 8; ++r) {
                const int  lrow = mt * 16 + r + hi8;
                const long grow = wgBase + lrow;
                const int  p    = lds_pid[lrow];
                const bool sel  = KNOWN ? (p != 0) : (p == 0);
                if (grow < N_ROWS && sel)
                    out[(size_t)grow * HID + colB] = acc[mt][r] + bias;
            }
        }
    }
}

extern "C" __global__ __launch_bounds__(BLOCK, 1)
void hpe_fused_kernel(const int* __restrict__ player_ids, const int* __restrict__ team_ids,
                      const int* __restrict__ role_ids,
                      const float* __restrict__ player_tab, const float* __restrict__ team_tab,
                      const float* __restrict__ role_tab,
                      const float* __restrict__ W1p, const float* __restrict__ b1p,
                      const float* __restrict__ gp,  const float* __restrict__ bp,
                      const float* __restrict__ W2p, const float* __restrict__ b2p,
                      const float* __restrict__ W1f, const float* __restrict__ b1f,
                      const float* __restrict__ gf,  const float* __restrict__ bf_,
                      const float* __restrict__ W2f, const float* __restrict__ b2f,
                      float* __restrict__ out)
{
    extern __shared__ char smem[];
    float* lds_h     = (float*)smem;                  // MT*HSTRIDE f32
    float* lds_x     = lds_h + MT * HSTRIDE;          // MT*XSTRIDE f32
    float* lds_stats = lds_x + MT * XSTRIDE;          // MT*2 f32
    int*   lds_pid   = (int*)(lds_stats + MT * 2);    // MT
    int*   lds_tid   = lds_pid + MT;                  // MT
    int*   lds_rid   = lds_tid + MT;                  // MT

    const int  tid    = threadIdx.x;
    const long wgBase = (long)blockIdx.x * MT;

    // stage ids (invalid tail rows -> id 0, stores are guarded by row < N)
    for (int r = tid; r < MT; r += BLOCK) {
        const long gr = wgBase + r;
        int p = 0, t = 0, ro = 0;
        if (gr < N_ROWS) { p = player_ids[gr]; t = team_ids[gr]; ro = role_ids[gr]; }
        lds_pid[r] = p; lds_tid[r] = t; lds_rid[r] = ro;
    }
    __syncthreads();

    // ---- known path: gather player embeddings (48 x 256, float4) ----
    for (int i = tid; i < MT * 64; i += BLOCK) {
        const int r = i >> 6, c4 = i & 63;
        const float4 v = ((const float4*)(player_tab + (size_t)lds_pid[r] * 256))[c4];
        *(float4*)(lds_x + r * XSTRIDE + c4 * 4) = v;
    }
    __syncthreads();
    run_path<256, true>(lds_h, lds_x, lds_stats, lds_pid,
                        W1p, b1p, gp, bp, W2p, b2p, out, wgBase);
    __syncthreads();

    // ---- fallback path: [team(128) | role(64)] = 48 x 192 ----
    for (int i = tid; i < MT * 48; i += BLOCK) {
        const int r  = i / 48;
        const int c4 = i - r * 48;
        float4 v;
        if (c4 < 32) v = ((const float4*)(team_tab + (size_t)lds_tid[r] * 128))[c4];
        else         v = ((const float4*)(role_tab + (size_t)lds_rid[r] * 64))[c4 - 32];
        *(float4*)(lds_x + r * XSTRIDE + c4 * 4) = v;
    }
    __syncthreads();
    run_path<192, false>(lds_h, lds_x, lds_stats, lds_pid,
                         W1f, b1f, gf, bf_, W2f, b2f, out, wgBase);
}

extern "C" void kernel_launch(void* const* d_in, const int* in_sizes, int n_in,
                              void* d_out, int out_size, void* d_ws, size_t ws_size,
                              hipStream_t stream) {
    (void)in_sizes; (void)n_in; (void)out_size; (void)d_ws; (void)ws_size;

    const int*   player_ids = (const int*)  d_in[0];
    const int*   team_ids   = (const int*)  d_in[1];
    const int*   role_ids   = (const int*)  d_in[2];
    const float* player_tab = (const float*)d_in[3];
    const float* team_tab   = (const float*)d_in[4];
    const float* role_tab   = (const float*)d_in[5];
    const float* W1p        = (const float*)d_in[6];
    const float* b1p        = (const float*)d_in[7];
    const float* gp         = (const float*)d_in[8];
    const float* bp         = (const float*)d_in[9];
    const float* W2p        = (const float*)d_in[10];
    const float* b2p        = (const float*)d_in[11];
    const float* W1f        = (const float*)d_in[12];
    const float* b1f        = (const float*)d_in[13];
    const float* gf         = (const float*)d_in[14];
    const float* bf_        = (const float*)d_in[15];
    const float* W2f        = (const float*)d_in[16];
    const float* b2f        = (const float*)d_in[17];
    float* out = (float*)d_out;

    const size_t smem_bytes = (size_t)(MT * HSTRIDE + MT * XSTRIDE + MT * 2) * sizeof(float)
                            + (size_t)(3 * MT) * sizeof(int);  // ~243 KB (WGP has 320 KB)

    (void)hipFuncSetAttribute((const void*)hpe_fused_kernel,
                              hipFuncAttributeMaxDynamicSharedMemorySize,
                              (int)smem_bytes);

    dim3 grid((N_ROWS + MT - 1) / MT);  // 2731 workgroups
    hpe_fused_kernel<<<grid, BLOCK, smem_bytes, stream>>>(
        player_ids, team_ids, role_ids, player_tab, team_tab, role_tab,
        W1p, b1p, gp, bp, W2p, b2p, W1f, b1f, gf, bf_, W2f, b2f, out);
}